// ConvNetWithGlobalPooling_52639119179977
// MI455X (gfx1250) — compile-verified
//
#include <hip/hip_runtime.h>

typedef float v2f __attribute__((ext_vector_type(2)));
typedef float v8f __attribute__((ext_vector_type(8)));

// Padded spatial layout: 66x66 with a 1-pixel zero halo around the 64x64 image.
// Activations are channel-PAIR interleaved: [B][C/2][66][66][2] so that the
// (even,odd) channel values of one pixel form one 8-byte unit == one WMMA
// B-fragment VGPR pair.
#define PS  66
#define PCH (66 * 66)

// ---------------------------------------------------------------------------
// Pad + interleave x: [32][3][64][64] -> xpi: [32][2][66][66][2]
// (channel 3 == 0, halo == 0)
// ---------------------------------------------------------------------------
__global__ __launch_bounds__(256) void pad_input_kernel(
    const float* __restrict__ x, float* __restrict__ xpi) {
  int i = blockIdx.x * blockDim.x + threadIdx.x;
  const int total = 32 * 4 * PCH;  // floats
  if (i >= total) return;
  int c2 = i & 1;
  int q  = i >> 1;
  int xx = q % PS;  q /= PS;
  int yy = q % PS;  q /= PS;
  int pr = q % 2;   // channel pair
  int b  = q / 2;
  int c  = pr * 2 + c2;
  float v = 0.0f;
  if (c < 3 && yy >= 1 && yy <= 64 && xx >= 1 && xx <= 64)
    v = x[((b * 3 + c) * 64 + (yy - 1)) * 64 + (xx - 1)];
  xpi[i] = v;
}

// ---------------------------------------------------------------------------
// Repack weights [B][COUT][CIN][3][3] -> [B][COUT][9][CINP] (channel-minor,
// zero padded CIN -> CINP) so an A-fragment is one contiguous b64.
// ---------------------------------------------------------------------------
template <int CIN, int CINP, int COUT>
__global__ __launch_bounds__(256) void repack_w_kernel(
    const float* __restrict__ w, float* __restrict__ wr) {
  int i = blockIdx.x * blockDim.x + threadIdx.x;
  const int total = 32 * COUT * 9 * CINP;
  if (i >= total) return;
  int ci = i % CINP;
  int q  = i / CINP;
  int t  = q % 9;  q /= 9;
  int co = q % COUT;
  int b  = q / COUT;
  wr[i] = (ci < CIN) ? w[((size_t)(b * COUT + co) * CIN + ci) * 9 + t] : 0.0f;
}

// ---------------------------------------------------------------------------
// Per-batch-weight 3x3 conv (SAME) as implicit GEMM on V_WMMA_F32_16X16X4_F32.
//   in  : [B][CINP/2][66][66][2]  zero-haloed, pair-interleaved activations
//   w   : [B][COUT][9][CINP]      repacked weights (channel-minor)
//   bias: [B][COUT]
//   out : OUTPAD ? [B][COUT/2][66][66][2] : [B][COUT/2][64][64][2]
// One wave owns a 32(Cout) x 64(one image row) tile: 2 co-tiles x 4 n-tiles.
// Every fragment is ONE aligned b64 load; every store is a b64 pair.
// ---------------------------------------------------------------------------
template <int CINP, int COUT, bool OUTPAD>
__global__ __launch_bounds__(256) void conv3x3_wmma_f32(
    const float* __restrict__ in, const float* __restrict__ w,
    const float* __restrict__ bias, float* __restrict__ out) {
  const int lane  = threadIdx.x & 31;
  const int wave  = threadIdx.x >> 5;
  const int b     = blockIdx.z;
  const int y     = blockIdx.y * 8 + wave;  // image row 0..63
  const int co0   = blockIdx.x * 32;
  const int n     = lane & 15;              // A-row (M) == B-col (N) index
  const int khalf = (lane >> 4) << 1;       // lanes 0-15 -> K{0,1}; 16-31 -> K{2,3}

  const float* __restrict__ inB = in + (size_t)b * (CINP / 2) * PCH * 2;
  const float* __restrict__ wA  = w + ((size_t)b * COUT + co0 + n) * (9 * CINP);
  constexpr int WTILE = 16 * 9 * CINP;      // co-tile stride in repacked weights

  v8f acc[2][4];
#pragma unroll
  for (int mt = 0; mt < 2; ++mt)
#pragma unroll
    for (int nt = 0; nt < 4; ++nt) acc[mt][nt] = (v8f)0.0f;

#pragma unroll
  for (int r = 0; r < 3; ++r) {
    // interleaved padded row (y-1+r)+1 = y+r
    const float* __restrict__ inRow = inB + (size_t)(y + r) * PS * 2;
#pragma unroll
    for (int s = 0; s < 3; ++s) {
      const float* __restrict__ wRS = wA + (r * 3 + s) * CINP;
#pragma unroll 2
      for (int ci = 0; ci < CINP; ci += 4) {
        const int k0 = ci + khalf;  // even
        // ---- A fragments: channels (k0,k0+1) adjacent -> single b64 each
        const v2f a0 = *(const v2f*)(wRS + k0);
        const v2f a1 = *(const v2f*)(wRS + k0 + WTILE);
        // ---- B base: pair plane k0/2, pixel (y+r-1, x+s-1) in padded coords
        const float* __restrict__ pB =
            inRow + (size_t)(k0 >> 1) * (PCH * 2) + (n + s) * 2;
#pragma unroll
        for (int nt = 0; nt < 4; ++nt) {
          const v2f bb = *(const v2f*)(pB + nt * 32);  // one b64
          acc[0][nt] = __builtin_amdgcn_wmma_f32_16x16x4_f32(
              false, a0, false, bb, (short)0, acc[0][nt], false, false);
          acc[1][nt] = __builtin_amdgcn_wmma_f32_16x16x4_f32(
              false, a1, false, bb, (short)0, acc[1][nt], false, false);
        }
      }
    }
  }

  // Epilogue: bias + ReLU + paired b64 stores.
  // C/D layout: VGPR v -> M = v + 8*laneHalf; v and v+1 are consecutive co.
  const int mBase = (lane >> 4) << 3;
#pragma unroll
  for (int mt = 0; mt < 2; ++mt) {
#pragma unroll
    for (int v = 0; v < 8; v += 2) {
      const int   co  = co0 + mt * 16 + mBase + v;  // even
      const float bi0 = bias[b * COUT + co];
      const float bi1 = bias[b * COUT + co + 1];
      float* __restrict__ o =
          OUTPAD
              ? out + (((size_t)b * (COUT / 2) + (co >> 1)) * PCH +
                       (y + 1) * PS + 1 + n) * 2
              : out + (((size_t)b * (COUT / 2) + (co >> 1)) * 4096 +
                       y * 64 + n) * 2;
#pragma unroll
      for (int nt = 0; nt < 4; ++nt) {
        v2f val;
        float t0 = acc[mt][nt][v] + bi0;
        float t1 = acc[mt][nt][v + 1] + bi1;
        val.x = t0 > 0.0f ? t0 : 0.0f;
        val.y = t1 > 0.0f ? t1 : 0.0f;
        *(v2f*)(o + nt * 32) = val;
      }
    }
  }
}

// ---------------------------------------------------------------------------
// Global mean-pool over 64x64 + rank-1 FC on pair-interleaved h3:
//   h3 : [32][128][64*64][2]
//   out[b][n][o] = mean_hw(h3[b][n]) * fcw[b][o] + bias4[b][o]
// One 256-thread block per (batch, channel-pair); emits 2 channels x 10.
// ---------------------------------------------------------------------------
__global__ __launch_bounds__(256) void pool_fc_kernel(
    const float* __restrict__ h3, const float* __restrict__ fcw,
    const float* __restrict__ bias4, float* __restrict__ out) {
  const int cp = blockIdx.x;  // channel pair 0..127
  const int b  = blockIdx.y;  // batch 0..31
  const float* __restrict__ p = h3 + ((size_t)b * 128 + cp) * 4096 * 2;

  float s0 = 0.0f, s1 = 0.0f;
#pragma unroll 4
  for (int i = threadIdx.x; i < 4096; i += 256) {
    v2f v = *(const v2f*)(p + i * 2);
    s0 += v.x;
    s1 += v.y;
  }

  __shared__ float red0[256];
  __shared__ float red1[256];
  red0[threadIdx.x] = s0;
  red1[threadIdx.x] = s1;
  __syncthreads();
#pragma unroll
  for (int off = 128; off > 0; off >>= 1) {
    if (threadIdx.x < off) {
      red0[threadIdx.x] += red0[threadIdx.x + off];
      red1[threadIdx.x] += red1[threadIdx.x + off];
    }
    __syncthreads();
  }
  const float pool0 = red0[0] * (1.0f / 4096.0f);
  const float pool1 = red1[0] * (1.0f / 4096.0f);
  if (threadIdx.x < 10) {
    const int o = threadIdx.x;
    const float fw = fcw[b * 10 + o];
    const float bs = bias4[b * 10 + o];
    out[((size_t)b * 256 + 2 * cp) * 10 + o]     = pool0 * fw + bs;
    out[((size_t)b * 256 + 2 * cp + 1) * 10 + o] = pool1 * fw + bs;
  }
}

extern "C" void kernel_launch(void* const* d_in, const int* in_sizes, int n_in,
                              void* d_out, int out_size, void* d_ws,
                              size_t ws_size, hipStream_t stream) {
  (void)in_sizes; (void)n_in; (void)out_size; (void)ws_size;
  const float* x   = (const float*)d_in[0];  // [32][3][64][64]
  const float* w1  = (const float*)d_in[1];  // [32][64][3][3][3]
  const float* w2  = (const float*)d_in[2];  // [32][128][64][3][3]
  const float* w3  = (const float*)d_in[3];  // [32][256][128][3][3]
  const float* fcw = (const float*)d_in[4];  // [32][1][10]
  const float* b1  = (const float*)d_in[5];  // [32][64]
  const float* b2  = (const float*)d_in[6];  // [32][128]
  const float* b3  = (const float*)d_in[7];  // [32][256]
  const float* b4  = (const float*)d_in[8];  // [32][10]
  float* out = (float*)d_out;                // [32][256][10]

  char* ws = (char*)d_ws;
  const size_t XPI_B = (size_t)32 * 4 * PCH * 4;          //  2,230,272
  const size_t W1R_B = (size_t)32 * 64 * 9 * 4 * 4;       //    294,912
  const size_t W2R_B = (size_t)32 * 128 * 9 * 64 * 4;     //  9,437,184
  const size_t W3R_B = (size_t)32 * 256 * 9 * 128 * 4;    // 37,748,736
  const size_t H1P_B = (size_t)32 * 64 * PCH * 4;         // 35,684,352
  const size_t H2P_B = (size_t)32 * 128 * PCH * 4;        // 71,368,704

  float* xpi = (float*)(ws);
  float* w1r = (float*)(ws + XPI_B);
  float* w2r = (float*)(ws + XPI_B + W1R_B);
  float* w3r = (float*)(ws + XPI_B + W1R_B + W2R_B);
  float* h1p = (float*)(ws + XPI_B + W1R_B + W2R_B + W3R_B);
  float* h2p = (float*)(ws + XPI_B + W1R_B + W2R_B + W3R_B + H1P_B);
  float* h3  = (float*)(ws + XPI_B + W1R_B + W2R_B + W3R_B + H1P_B + H2P_B);

  // Stage 0: halo zeroing + layout transforms (all stream-ordered).
  hipMemsetAsync(h1p, 0, H1P_B, stream);  // halos; interiors overwritten
  hipMemsetAsync(h2p, 0, H2P_B, stream);
  {
    const int tX = 32 * 4 * PCH;
    pad_input_kernel<<<(tX + 255) / 256, 256, 0, stream>>>(x, xpi);
    const int t1 = 32 * 64 * 9 * 4;
    repack_w_kernel<3, 4, 64>
        <<<(t1 + 255) / 256, 256, 0, stream>>>(w1, w1r);
    const int t2 = 32 * 128 * 9 * 64;
    repack_w_kernel<64, 64, 128>
        <<<(t2 + 255) / 256, 256, 0, stream>>>(w2, w2r);
    const int t3 = 32 * 256 * 9 * 128;
    repack_w_kernel<128, 128, 256>
        <<<(t3 + 255) / 256, 256, 0, stream>>>(w3, w3r);
  }

  // grid = (Cout/32, 64/8 row groups, batch); block = 256 threads = 8 waves.
  conv3x3_wmma_f32<4, 64, true>
      <<<dim3(2, 8, 32), 256, 0, stream>>>(xpi, w1r, b1, h1p);
  conv3x3_wmma_f32<64, 128, true>
      <<<dim3(4, 8, 32), 256, 0, stream>>>(h1p, w2r, b2, h2p);
  conv3x3_wmma_f32<128, 256, false>
      <<<dim3(8, 8, 32), 256, 0, stream>>>(h2p, w3r, b3, h3);
  pool_fc_kernel<<<dim3(128, 32), 256, 0, stream>>>(h3, fcw, b4, out);
}